// SpectralConv2d_72533407695317
// MI455X (gfx1250) — compile-verified
//
#include <hip/hip_runtime.h>
#include <hip/hip_bf16.h>
#include <math.h>

// ---------------------------------------------------------------------------
// SpectralConv2d on MI455X (gfx1250), wave32 + V_WMMA_F32_16X16X4_F32.
//
// B=16, C=64, H=256, W=256, M1=M2=32.
// Per (b,c) image: partial DFT chain implemented as f32 WMMA GEMMs.
//   T1: Y = X * Fw            (256x256)x(256x32)  cplx result
//   T2: Xf = Fh * Y           (64x256)x(256x32)   cplx
//   T3: Of = Xf .* wsum(c)    pointwise complex
//   T4: Z = Gh * Of           (256x64)x(64x32)    cplx
//   T5: out = Zr*Cr + Zi*Ci   (256x32)x(32x256)   real
//
// Twiddle tables are generated on-device into d_ws, pre-packed in WMMA
// fragment layout (one float2 per lane), L2-resident, reused by all blocks.
// ---------------------------------------------------------------------------

typedef float v2f __attribute__((ext_vector_type(2)));
typedef float v8f __attribute__((ext_vector_type(8)));

#define WMMA4(c, a, b)                                                        \
  (c) = __builtin_amdgcn_wmma_f32_16x16x4_f32(false, (a), false, (b),         \
                                              (short)0, (c), false, false)

// ---- workspace layout (in floats) -----------------------------------------
#define WS_W    0         // summed weights: [c][p][k2][{r,i}]  64*64*32*2
#define WS_F1C  262144    // T1 B-frags cos   (64 kb * 2 ng * 32 * 2)
#define WS_F1S  270336    // T1 B-frags -sin
#define WS_C2   278528    // T2 A-frags cos   (4 mg * 64 kb * 32 * 2)
#define WS_S2   294912    // T2 A-frags +sin
#define WS_S2N  311296    // T2 A-frags -sin
#define WS_C4   327680    // T4 A-frags cos   (16 mg * 16 kb * 32 * 2)
#define WS_S4   344064    // T4 A-frags +sin
#define WS_S4N  360448    // T4 A-frags -sin
#define WS_CR   376832    // T5 B-frags  a*cos/65536  (8 kb * 16 ng * 32 * 2)
#define WS_CI   385024    // T5 B-frags -a*sin/65536
#define WS_END  393216    // 1.57 MB total

// ---------------------------------------------------------------------------
// Reduce weights over the in-channel axis j:
//   wsum[c][p][k2] = sum_j w{1|2}[j][c][m1][k2]  (p<32 -> w1, p>=32 -> w2)
// ---------------------------------------------------------------------------
__global__ __launch_bounds__(256) void init_wsum(
    const float* __restrict__ w1r, const float* __restrict__ w1i,
    const float* __restrict__ w2r, const float* __restrict__ w2i,
    float* __restrict__ ws) {
  int i = blockIdx.x * 256 + threadIdx.x;  // 0 .. 131071
  int c  = i >> 11;
  int r  = i & 2047;   // p*32 + k2
  int p  = r >> 5;
  int m1 = p & 31;
  int k2 = r & 31;
  const float* wr = (p < 32) ? w1r : w2r;
  const float* wi = (p < 32) ? w1i : w2i;
  int base = c * 1024 + m1 * 32 + k2;      // within one j-slice (C*M1*M2)
  float sr = 0.f, si = 0.f;
  for (int j = 0; j < 64; ++j) {
    sr += wr[j * 65536 + base];
    si += wi[j * 65536 + base];
  }
  ws[WS_W + 2 * (c * 2048 + r) + 0] = sr;
  ws[WS_W + 2 * (c * 2048 + r) + 1] = si;
}

// cos/sin of 2*pi*m/256 with exact integer phase reduction
__device__ inline float2 cs256(int m) {
  float th = (float)(m & 255) * (6.28318530717958647692f / 256.0f);
  float s, c;
  __sincosf(th, &s, &c);
  return make_float2(c, s);
}

// ---------------------------------------------------------------------------
// Generate all twiddle tables in WMMA fragment layout.
// One thread per float2 entry; 65536 entries total.
//   A-frag: lane (l<16): rows 16mg+l, cols 4kb+{0,1}; lane>=16: cols 4kb+{2,3}
//   B-frag: lane (l<16): rows 4kb+{0,1}, col 16ng+l; lane>=16: rows 4kb+{2,3}
// ---------------------------------------------------------------------------
__global__ __launch_bounds__(256) void init_twiddles(float* __restrict__ ws) {
  int t = blockIdx.x * 256 + threadIdx.x;  // 0 .. 65535
  int lane = t & 31;
  int lrow = lane & 15;
  int lhi  = lane >> 4;
  float v0 = 0.f, v1 = 0.f;
  int off = 0;

  if (t < 8192) {
    // F1c / F1s : B[w][k2] = exp(-2pi i * w*k2/256) for T1
    int e  = t & 4095;
    bool isS = (t >= 4096);
    int kb = e >> 6;
    int ng = (e >> 5) & 1;
    int k  = kb * 4 + lhi * 2;       // w row
    int n  = ng * 16 + lrow;         // k2 col
    float2 a = cs256(k * n), b = cs256((k + 1) * n);
    v0 = isS ? -a.y : a.x;
    v1 = isS ? -b.y : b.x;
    off = WS_F1C + t * 2;
  } else if (t < 32768) {
    // C2/S2/S2N : A[p][h], theta = 2pi*k1(p)*h/256 (forward H DFT)
    int u = t - 8192;
    int e = u & 8191;
    int which = u >> 13;             // 0=cos 1=+sin 2=-sin
    int kb = (e >> 5) & 63;
    int mg = e >> 11;
    int p  = mg * 16 + lrow;
    int k1 = (p < 32) ? p : p + 192;
    int h0 = kb * 4 + lhi * 2;
    float2 a = cs256(k1 * h0), b = cs256(k1 * (h0 + 1));
    v0 = (which == 0) ? a.x : ((which == 1) ? a.y : -a.y);
    v1 = (which == 0) ? b.x : ((which == 1) ? b.y : -b.y);
    off = WS_C2 + u * 2;
  } else if (t < 57344) {
    // C4/S4/S4N : A[h][p], phi = 2pi*k1(p)*h/256 (inverse H DFT)
    int u = t - 32768;
    int e = u & 8191;
    int which = u >> 13;
    int kb = (e >> 5) & 15;
    int mg = e >> 9;
    int h  = mg * 16 + lrow;
    int p0 = kb * 4 + lhi * 2;
    int k10 = (p0 < 32) ? p0 : p0 + 192;
    int p1 = p0 + 1;
    int k11 = (p1 < 32) ? p1 : p1 + 192;
    float2 a = cs256(k10 * h), b = cs256(k11 * h);
    v0 = (which == 0) ? a.x : ((which == 1) ? a.y : -a.y);
    v1 = (which == 0) ? b.x : ((which == 1) ? b.y : -b.y);
    off = WS_C4 + u * 2;
  } else {
    // CR/CI : B[k2][w] = a_{k2} * {cos,-sin}(2pi*k2*w/256) / (H*W)
    int u = t - 57344;
    int e = u & 4095;
    bool isI = (t >= 61440);
    int ng = (e >> 5) & 15;
    int kb = e >> 9;
    int k20 = kb * 4 + lhi * 2;
    int w  = ng * 16 + lrow;
    float sc0 = ((k20 == 0) ? 1.f : 2.f) * (1.f / 65536.f);
    float sc1 = 2.f * (1.f / 65536.f);  // k20+1 is never 0
    float2 a = cs256(k20 * w), b = cs256((k20 + 1) * w);
    v0 = (isI ? -a.y : a.x) * sc0;
    v1 = (isI ? -b.y : b.x) * sc1;
    off = WS_CR + u * 2;
  }
  ws[off] = v0;
  ws[off + 1] = v1;
}

// ---------------------------------------------------------------------------
// Main kernel: one workgroup (8 wave32) per (b,c) image.
// LDS: Yr/Yi 256x32 f32 (64KB, aliased by Zr/Zi) + Xfr/Xfi 64x32 (16KB).
// ---------------------------------------------------------------------------
__global__ __launch_bounds__(256) void spectral_main(
    const float* __restrict__ x, const float* __restrict__ ws,
    float* __restrict__ out) {
  extern __shared__ float smem[];
  float* Yr  = smem;          // 8192 floats (aliased as Zr in T4/T5)
  float* Yi  = smem + 8192;   // 8192 floats (aliased as Zi)
  float* Xfr = smem + 16384;  // 2048 floats
  float* Xfi = smem + 18432;  // 2048 floats

  const int bc   = blockIdx.x;       // b*64 + c
  const int c    = bc & 63;
  const int lane = threadIdx.x & 31;
  const int wave = threadIdx.x >> 5;
  const int lrow = lane & 15;
  const int lhi  = lane >> 4;

  const float* F1C = ws + WS_F1C;
  const float* F1S = ws + WS_F1S;
  const float* C2  = ws + WS_C2;
  const float* S2  = ws + WS_S2;
  const float* S2N = ws + WS_S2N;
  const float* C4  = ws + WS_C4;
  const float* S4  = ws + WS_S4;
  const float* S4N = ws + WS_S4N;
  const float* CR  = ws + WS_CR;
  const float* CI  = ws + WS_CI;

  const float* xim = x + (size_t)bc * 65536;
  float* oim = out + (size_t)bc * 65536;

  // ---- T1: Y = X * Fw  (each wave: 2 row-strips of 16, full N=32, r&i) ----
  for (int s = wave; s < 16; s += 8) {
    v8f yr0 = {}, yr1 = {}, yi0 = {}, yi1 = {};
    const float* arow = xim + (s * 16 + lrow) * 256 + lhi * 2;
    for (int kb = 0; kb < 64; ++kb) {
      v2f a = *(const v2f*)(arow + kb * 4);
      int fb = (kb * 2) * 32 + lane;
      v2f bc0 = *(const v2f*)(F1C + 2 * fb);
      v2f bc1 = *(const v2f*)(F1C + 2 * (fb + 32));
      v2f bs0 = *(const v2f*)(F1S + 2 * fb);
      v2f bs1 = *(const v2f*)(F1S + 2 * (fb + 32));
      WMMA4(yr0, a, bc0);
      WMMA4(yr1, a, bc1);
      WMMA4(yi0, a, bs0);
      WMMA4(yi1, a, bs1);
    }
#pragma unroll
    for (int v = 0; v < 8; ++v) {
      int row = s * 16 + v + lhi * 8;
      Yr[row * 32 + lrow]      = yr0[v];
      Yr[row * 32 + 16 + lrow] = yr1[v];
      Yi[row * 32 + lrow]      = yi0[v];
      Yi[row * 32 + 16 + lrow] = yi1[v];
    }
  }
  __syncthreads();

  // ---- T2: Xf = Fh * Y (complex; 8 (mg,ng) tiles -> one per wave) --------
  {
    const int mg = wave >> 1, ng = wave & 1;
    const int colc = ng * 16 + lrow;
    v8f xr = {}, xi = {};
    for (int kb = 0; kb < 64; ++kb) {
      int fa = (mg * 64 + kb) * 32 + lane;
      v2f ac  = *(const v2f*)(C2 + 2 * fa);
      v2f as  = *(const v2f*)(S2 + 2 * fa);
      v2f asn = *(const v2f*)(S2N + 2 * fa);
      int k = kb * 4 + lhi * 2;
      v2f byr, byi;
      byr.x = Yr[k * 32 + colc];       byr.y = Yr[(k + 1) * 32 + colc];
      byi.x = Yi[k * 32 + colc];       byi.y = Yi[(k + 1) * 32 + colc];
      WMMA4(xr, ac, byr);   // cos*Yr
      WMMA4(xr, as, byi);   // + sin*Yi
      WMMA4(xi, ac, byi);   // cos*Yi
      WMMA4(xi, asn, byr);  // - sin*Yr
    }
#pragma unroll
    for (int v = 0; v < 8; ++v) {
      int p = mg * 16 + v + lhi * 8;
      Xfr[p * 32 + colc] = xr[v];
      Xfi[p * 32 + colc] = xi[v];
    }
  }
  __syncthreads();

  // ---- T3: Of = Xf .* wsum(c)  (pointwise complex, in place) -------------
  for (int i = threadIdx.x; i < 2048; i += 256) {
    float ar = Xfr[i], ai = Xfi[i];
    const float* wp = ws + WS_W + (size_t)(c * 2048 + i) * 2;
    float wr = wp[0], wi = wp[1];
    Xfr[i] = ar * wr - ai * wi;
    Xfi[i] = ar * wi + ai * wr;
  }
  __syncthreads();

  // ---- T4: Z = Gh * Of (complex; 32 (mg,ng) tiles -> 4 per wave) ---------
  for (int j = 0; j < 4; ++j) {
    const int q = wave * 4 + j;        // 0..31
    const int mg = q >> 1, ng = q & 1;
    const int colc = ng * 16 + lrow;
    v8f zr = {}, zi = {};
    for (int kb = 0; kb < 16; ++kb) {
      int fa = (mg * 16 + kb) * 32 + lane;
      v2f ac  = *(const v2f*)(C4 + 2 * fa);
      v2f as  = *(const v2f*)(S4 + 2 * fa);
      v2f asn = *(const v2f*)(S4N + 2 * fa);
      int k = kb * 4 + lhi * 2;
      v2f br, bi;
      br.x = Xfr[k * 32 + colc];       br.y = Xfr[(k + 1) * 32 + colc];
      bi.x = Xfi[k * 32 + colc];       bi.y = Xfi[(k + 1) * 32 + colc];
      WMMA4(zr, ac, br);    // cos*Ofr
      WMMA4(zr, asn, bi);   // - sin*Ofi
      WMMA4(zi, ac, bi);    // cos*Ofi
      WMMA4(zi, as, br);    // + sin*Ofr
    }
#pragma unroll
    for (int v = 0; v < 8; ++v) {
      int h = mg * 16 + v + lhi * 8;
      Yr[h * 32 + colc] = zr[v];   // Zr aliases Yr
      Yi[h * 32 + colc] = zi[v];   // Zi aliases Yi
    }
  }
  __syncthreads();

  // ---- T5: out = Zr*Cr + Zi*Ci  (256 tiles -> 32 per wave) ---------------
  for (int j = 0; j < 32; ++j) {
    const int tile = wave * 32 + j;
    const int mg = tile >> 4, ng = tile & 15;
    const int row = mg * 16 + lrow;
    v8f o = {};
    for (int kb = 0; kb < 8; ++kb) {
      int k = kb * 4 + lhi * 2;
      v2f azr = *(const v2f*)(&Yr[row * 32 + k]);
      v2f azi = *(const v2f*)(&Yi[row * 32 + k]);
      int fb = (kb * 16 + ng) * 32 + lane;
      v2f bcr = *(const v2f*)(CR + 2 * fb);
      v2f bci = *(const v2f*)(CI + 2 * fb);
      WMMA4(o, azr, bcr);
      WMMA4(o, azi, bci);
    }
#pragma unroll
    for (int v = 0; v < 8; ++v) {
      oim[(mg * 16 + v + lhi * 8) * 256 + ng * 16 + lrow] = o[v];
    }
  }
}

// ---------------------------------------------------------------------------
extern "C" void kernel_launch(void* const* d_in, const int* in_sizes, int n_in,
                              void* d_out, int out_size, void* d_ws,
                              size_t ws_size, hipStream_t stream) {
  const float* x   = (const float*)d_in[0];
  const float* w1r = (const float*)d_in[1];
  const float* w1i = (const float*)d_in[2];
  const float* w2r = (const float*)d_in[3];
  const float* w2i = (const float*)d_in[4];
  float* ws  = (float*)d_ws;   // needs >= WS_END*4 = ~1.6 MB
  float* out = (float*)d_out;

  // 1) reduce weights over in-channel axis
  init_wsum<<<512, 256, 0, stream>>>(w1r, w1i, w2r, w2i, ws);
  // 2) build fragment-packed DFT twiddle tables
  init_twiddles<<<256, 256, 0, stream>>>(ws);
  // 3) spectral conv: one workgroup per (b,c) image, 80 KB dynamic LDS
  spectral_main<<<1024, 256, 80 * 1024, stream>>>(x, ws, out);
}